// YOLOv3_Actuator_30425548325118
// MI455X (gfx1250) — compile-verified
//
#include <hip/hip_runtime.h>

typedef __attribute__((ext_vector_type(2))) float v2f;
typedef __attribute__((ext_vector_type(8))) float v8f;

#define NOOBJ_SCALE 0.5f
#define OBJ_SCALE   5.0f
#define NCLS 3
#define MAXNB 16

__device__ __forceinline__ float safe_log(float p) {
    return fmaxf(__logf(p), -100.0f);
}
__device__ __forceinline__ float bce(float p, float t) {
    return -(t * safe_log(p) + (1.0f - t) * safe_log(1.0f - p));
}

__global__ void init_acc_kernel(float* acc) {
    acc[0] = 0.0f;
}

// Dense pass: every cell contributes NOOBJ_SCALE * bce(pconf, 0) assuming
// obj=0, noobj=1. Sparse deviations are fixed up by the correction kernel.
// Wave reduction is done with V_WMMA_F32_16X16X4_F32 (A = ones) so the 32
// lane partials collapse on the matrix pipe.
__global__ __launch_bounds__(256) void bulk_noobj_kernel(
        const float* __restrict__ outp, long long ncells,
        float* __restrict__ acc) {
    long long idx    = (long long)blockIdx.x * blockDim.x + threadIdx.x;
    long long stride = (long long)gridDim.x * blockDim.x;
    float partial = 0.0f;
    for (long long c = idx; c < ncells; c += stride) {
        float p = outp[c * 8 + 4];                 // channel 4 = pconf
        partial += -fmaxf(__logf(1.0f - p), -100.0f);
    }
    partial *= NOOBJ_SCALE;

    // D = ones(16x4) x B(4x16) + 0 : every row of D = column sums of B.
    // Placing each lane's partial in one B slot (b[0]) and zero elsewhere
    // means sum over D row 0 == sum of all 32 partials, for ANY bijective
    // lane->slot layout.
    v2f a;  a[0] = 1.0f;     a[1] = 1.0f;
    v2f bm; bm[0] = partial; bm[1] = 0.0f;
    v8f cz = {};
    v8f d = __builtin_amdgcn_wmma_f32_16x16x4_f32(
        /*neg_a=*/false, a, /*neg_b=*/false, bm,
        /*c_mod=*/(short)0, cz, /*reuse_a=*/false, /*reuse_b=*/false);

    float s = d[0];                 // lane L holds colsum[n(L)], rows identical
    s += __shfl_xor(s, 1, 32);      // fold the 16 distinct columns
    s += __shfl_xor(s, 2, 32);      // (upper half mirrors lower half)
    s += __shfl_xor(s, 4, 32);
    s += __shfl_xor(s, 8, 32);

    __shared__ float wsum[8];
    int lane = threadIdx.x & 31;
    int wave = threadIdx.x >> 5;
    if (lane == 0) wsum[wave] = s;
    __syncthreads();
    if (threadIdx.x == 0) {
        float t = 0.0f;
        int nw = blockDim.x >> 5;
        for (int i = 0; i < nw; ++i) t += wsum[i];
        atomicAdd(acc, t);
    }
}

// Sparse pass: one thread per batch image (launched once per scale).
// Rebuilds the scatter targets exactly (max / min / last-writer .set /
// first-max argmax) and adds (actual cell loss - already-counted bulk term)
// for each affected (anchor, gj, gi) cell.
__global__ void yolo_correction_kernel(
        const float* __restrict__ outp,
        const float* __restrict__ boxes,
        const int*   __restrict__ labels,
        float* __restrict__ acc,
        int G, int B, int NB,
        float aw0, float ah0, float aw1, float ah1, float aw2, float ah2) {
    int b = blockIdx.x * blockDim.x + threadIdx.x;
    if (b >= B) return;

    float aw[3] = {aw0, aw1, aw2};
    float ah[3] = {ah0, ah1, ah2};

    int   gi[MAXNB], gj[MAXNB], best[MAXNB], lab[MAXNB];
    float vf[MAXNB], twv[MAXNB], thv[MAXNB], keep[MAXNB][3];

    for (int n = 0; n < NB; ++n) {
        const float* bx = boxes + ((size_t)b * NB + n) * 4;
        float tx = bx[0] * (float)G;
        float ty = bx[1] * (float)G;
        float bw = bx[2] * (float)G;
        float bh = bx[3] * (float)G;
        int ii = (int)tx; ii = ii < 0 ? 0 : (ii > G - 1 ? G - 1 : ii);
        int jj = (int)ty; jj = jj < 0 ? 0 : (jj > G - 1 ? G - 1 : jj);
        gi[n] = ii; gj[n] = jj;

        float iou[3]; float best_iou = -1.0f; int bi = 0;
        for (int a2 = 0; a2 < 3; ++a2) {
            float inter = fminf(aw[a2], bw) * fminf(ah[a2], bh);
            float uni   = aw[a2] * ah[a2] + 1e-16f + bw * bh - inter;
            iou[a2] = inter / uni;
            if (iou[a2] > best_iou) { best_iou = iou[a2]; bi = a2; }  // first max wins
        }
        best[n] = bi;
        int L = labels[(size_t)b * NB + n];
        bool valid = (L < NCLS);
        vf[n] = valid ? 1.0f : 0.0f;
        int Lc = L < 0 ? 0 : (L > NCLS - 1 ? NCLS - 1 : L);
        lab[n] = Lc;
        for (int a2 = 0; a2 < 3; ++a2)
            keep[n][a2] = (valid && (iou[a2] > 0.5f || a2 == bi)) ? 0.0f : 1.0f;
        twv[n] = vf[n] * __logf(bw / aw[bi] + 1e-16f);
        thv[n] = vf[n] * __logf(bh / ah[bi] + 1e-16f);
    }

    float lsum = 0.0f;
    for (int n = 0; n < NB; ++n) {
        bool dup = false;                       // handle each (gj,gi) once
        for (int m = 0; m < n; ++m)
            if (gi[m] == gi[n] && gj[m] == gj[n]) { dup = true; break; }
        if (dup) continue;

        for (int a2 = 0; a2 < 3; ++a2) {
            float obj = 0.0f, noobj = 1.0f, tw = 0.0f, th = 0.0f;
            float tcls[NCLS] = {0.0f, 0.0f, 0.0f};
            for (int m = 0; m < NB; ++m) {
                if (gi[m] != gi[n] || gj[m] != gj[n]) continue;
                noobj = fminf(noobj, keep[m][a2]);
                if (best[m] == a2) {
                    obj = fmaxf(obj, vf[m]);
                    tw = twv[m]; th = thv[m];            // .set: last writer wins
                    tcls[lab[m]] = fmaxf(tcls[lab[m]], vf[m]);
                }
            }
            size_t cidx = ((((size_t)b * 3 + a2) * G + gj[n]) * G + gi[n]) * 8;
            const float4* cp = (const float4*)(outp + cidx);
            float4 lo = cp[0];
            float4 hi = cp[1];
            float px = lo.x - floorf(lo.x);
            float py = lo.y - floorf(lo.y);
            float pw = __logf(lo.z / aw[a2] + 1e-16f);
            float ph = __logf(lo.w / ah[a2] + 1e-16f);
            float pconf = hi.x;
            float tconf = obj;
            float bconf = bce(pconf, tconf);
            float actual = obj * (px * px + py * py)
                         + obj * ((pw - tw) * (pw - tw) + (ph - th) * (ph - th))
                         + OBJ_SCALE * obj * bconf
                         + NOOBJ_SCALE * noobj * bconf
                         + obj * (bce(hi.y, tcls[0]) + bce(hi.z, tcls[1]) + bce(hi.w, tcls[2]));
            float bulk = NOOBJ_SCALE * bce(pconf, 0.0f);  // remove dense-pass term
            lsum += actual - bulk;
        }
    }
    atomicAdd(acc, lsum);
}

__global__ void finalize_kernel(const float* acc, float* out, float invB) {
    out[0] = acc[0] * invB;
}

extern "C" void kernel_launch(void* const* d_in, const int* in_sizes, int n_in,
                              void* d_out, int out_size, void* d_ws, size_t ws_size,
                              hipStream_t stream) {
    const float* outs[3] = { (const float*)d_in[0], (const float*)d_in[1], (const float*)d_in[2] };
    const float* boxes   = (const float*)d_in[3];
    const int*   labels  = (const int*)d_in[4];
    float* acc = (float*)d_ws;
    float* out = (float*)d_out;

    const int G[3] = {19, 38, 76};
    // SCALED_ANCHORS, row-major [scale][anchor][wh]
    const float anc[3][6] = {
        {116.0f/32.0f, 90.0f/32.0f, 156.0f/32.0f, 198.0f/32.0f, 373.0f/32.0f, 326.0f/32.0f},
        { 30.0f/16.0f, 61.0f/16.0f,  62.0f/16.0f,  45.0f/16.0f,  59.0f/16.0f, 119.0f/16.0f},
        { 10.0f/ 8.0f, 13.0f/ 8.0f,  16.0f/ 8.0f,  30.0f/ 8.0f,  33.0f/ 8.0f,  23.0f/ 8.0f}
    };

    int B  = in_sizes[0] / (3 * 19 * 19 * 8);
    if (B < 1) B = 1;
    int NB = in_sizes[4] / B;
    if (NB > MAXNB) NB = MAXNB;

    init_acc_kernel<<<1, 1, 0, stream>>>(acc);

    for (int s = 0; s < 3; ++s) {
        long long ncells = (long long)in_sizes[s] / 8;
        long long blk = (ncells + 255) / 256;
        if (blk > 4096) blk = 4096;
        if (blk < 1) blk = 1;
        bulk_noobj_kernel<<<(int)blk, 256, 0, stream>>>(outs[s], ncells, acc);
    }

    for (int s = 0; s < 3; ++s) {
        int tpb = 128;
        int nb  = (B + tpb - 1) / tpb;
        yolo_correction_kernel<<<nb, tpb, 0, stream>>>(
            outs[s], boxes, labels, acc, G[s], B, NB,
            anc[s][0], anc[s][1], anc[s][2], anc[s][3], anc[s][4], anc[s][5]);
    }

    finalize_kernel<<<1, 1, 0, stream>>>(acc, out, 1.0f / (float)B);
}